// MaskLayer_61211873902879
// MI455X (gfx1250) — compile-verified
//
#include <hip/hip_runtime.h>
#include <hip/hip_bf16.h>
#include <stdint.h>

// ---------------------------------------------------------------------------
// Fused thermal-mask pipeline for MI455X (gfx1250, wave32).
//   Pass 0: init workspace (min/max sentinels, histograms, thresholds)
//   Pass 1: thermal transform + separable 9-tap Gaussian blur via chained
//           V_WMMA_F32_16X16X4_F32, halo tiles staged with
//           global_load_async_to_lds_b32; per-image min/max atomics.
//   Pass 2: per-image 256-bin histogram (LDS-privatized).
//   Pass 3: per-image Otsu scan (one 256-thread block per image).
//   Pass 4: composite mask over RGB channels.
// Bandwidth-bound: ~268 MB DRAM traffic -> ~11.5 us at 23.3 TB/s. The 33.5 MB
// blurred intermediate lives in the 192 MB L2 between passes.
// ---------------------------------------------------------------------------

typedef __attribute__((ext_vector_type(2))) float v2f;
typedef __attribute__((ext_vector_type(8))) float v8f;

#define B       32
#define H_IMG   512
#define W_IMG   512
#define NPIX    (H_IMG * W_IMG)      // 262144
#define NBINS   256
#define RADIUS  4
#define HALO    24                   // 16 + 2*RADIUS
#define SPITCH  25                   // 24 padded -> gcd(25,64)=1, conflict-free
#define TPITCH  17

// Normalized Gaussian taps, sigma=1, radius=4 (sum = 1).
__constant__ float GW[9] = {
    1.3383062e-4f, 4.4318615e-3f, 5.3990967e-2f, 2.4197073e-1f,
    3.9894351e-1f, 2.4197073e-1f, 5.3990967e-2f, 4.4318615e-3f, 1.3383062e-4f};

__device__ __forceinline__ float gw(int d) {
    int i = min(max(d, 0), 8);
    return (d >= 0 && d <= 8) ? GW[i] : 0.0f;
}

__device__ __forceinline__ float thermal(float v) {
    return fminf(fmaxf(v * 0.5f + 0.5f, 0.0f), 1.0f);
}

__device__ __forceinline__ v8f wmma4(v2f a, v2f b, v8f c) {
    // D = A(16x4,f32) x B(4x16,f32) + C(16x16,f32)
    return __builtin_amdgcn_wmma_f32_16x16x4_f32(
        false, a, false, b, (short)0, c, false, false);
}

// ---------------------------------------------------------------------------
// Pass 0: init min/max sentinels, histograms, thresholds (ws is poisoned).
__global__ void init_kernel(int* minmax, unsigned* hist, float* thresh) {
    int i = blockIdx.x * blockDim.x + threadIdx.x;           // 8192 threads
    if (i < 2 * B) minmax[i] = (i & 1) ? 0 : 0x7F800000;     // min=+inf, max=0
    if (i < B * NBINS) hist[i] = 0u;
    if (i < B) thresh[i] = 0.0f;
}

// ---------------------------------------------------------------------------
// Pass 1: one wave (32 threads) per 16x16 output tile.
//   blurred(16x16) = V(16x24) * in(24x24) * H(24x16), K split into 6 WMMAs.
__global__ __launch_bounds__(32)
void blur_wmma_kernel(const float* __restrict__ x, float* __restrict__ blur,
                      int* __restrict__ minmax) {
    const int tile = blockIdx.x;          // 32*32*32 = 32768 tiles
    const int b    = tile >> 10;
    const int t2   = tile & 1023;
    const int y0   = (t2 >> 5) << 4;
    const int x0   = (t2 & 31) << 4;
    const int lane = threadIdx.x;
    const int half = lane >> 4;           // A/B K-split: lanes 0-15 -> K0,K1
    const int m    = lane & 15;

    __shared__ float sIn[HALO * SPITCH];  // raw x[:,0] halo tile
    __shared__ float sTmp[HALO * TPITCH]; // tmp = in * H  (24x16)

    const float* src = x + (size_t)b * 4 * NPIX;   // channel 0 (thermal)

    // --- Async halo load: global -> LDS, no VGPR staging (18 per lane) ---
    for (int i = lane; i < HALO * HALO; i += 32) {
        int r  = i / HALO, c = i - r * HALO;
        int gy = min(max(y0 - RADIUS + r, 0), H_IMG - 1);   // edge clamp
        int gx = min(max(x0 - RADIUS + c, 0), W_IMG - 1);
        const float* gaddr = src + gy * W_IMG + gx;
        unsigned ldsoff = (unsigned)(size_t)&sIn[r * SPITCH + c];
        asm volatile("global_load_async_to_lds_b32 %0, %1, off"
                     :: "v"(ldsoff), "v"(gaddr) : "memory");
    }
    asm volatile("s_wait_asynccnt 0x0" ::: "memory");
    __syncthreads();   // single-wave WG: lowers to S_NOP, compiler fence

    // H(24x16): H[r][c] = w[r-c]. B fragment chunk j: rows 4j+2*half+{0,1}.
    v2f hb[6];
#pragma unroll
    for (int j = 0; j < 6; ++j) {
        int r0 = 4 * j + 2 * half;
        hb[j].x = gw(r0 - m);
        hb[j].y = gw(r0 + 1 - m);
    }

    // Step 1: tmp = in * H, two 16-row blocks (rows 0-15 and 8-23).
    v8f t0 = {}, t1 = {};
#pragma unroll
    for (int j = 0; j < 6; ++j) {
        int c0 = 4 * j + 2 * half;
        v2f a0, a1;
        a0.x = thermal(sIn[m * SPITCH + c0]);
        a0.y = thermal(sIn[m * SPITCH + c0 + 1]);
        a1.x = thermal(sIn[(m + 8) * SPITCH + c0]);
        a1.y = thermal(sIn[(m + 8) * SPITCH + c0 + 1]);
        t0 = wmma4(a0, hb[j], t0);
        t1 = wmma4(a1, hb[j], t1);
    }

    // D layout: VGPR j -> row j + 8*half, col m. Re-layout via LDS for B use.
    // Rows 8-15 written twice with bitwise-identical values (benign).
#pragma unroll
    for (int j = 0; j < 8; ++j) {
        sTmp[(j + 8 * half) * TPITCH + m]     = t0[j];
        sTmp[(8 + j + 8 * half) * TPITCH + m] = t1[j];
    }
    __syncthreads();

    // Step 2: out = V * tmp, V[m][r] = w[r-m].
    v8f o = {};
#pragma unroll
    for (int j = 0; j < 6; ++j) {
        int k0 = 4 * j + 2 * half;
        v2f va, tb;
        va.x = gw(k0 - m);
        va.y = gw(k0 + 1 - m);
        tb.x = sTmp[k0 * TPITCH + m];
        tb.y = sTmp[(k0 + 1) * TPITCH + m];
        o = wmma4(va, tb, o);
    }

    // Store blurred tile + per-image min/max (values in [0,1] -> int order).
    float vmin = 3.4e38f, vmax = -3.4e38f;
    float* dst = blur + (size_t)b * NPIX;
#pragma unroll
    for (int j = 0; j < 8; ++j) {
        int row = y0 + j + 8 * half;
        float v = o[j];
        dst[row * W_IMG + x0 + m] = v;
        vmin = fminf(vmin, v);
        vmax = fmaxf(vmax, v);
    }
#pragma unroll
    for (int off = 16; off > 0; off >>= 1) {
        vmin = fminf(vmin, __shfl_xor(vmin, off));
        vmax = fmaxf(vmax, __shfl_xor(vmax, off));
    }
    if (lane == 0) {
        atomicMin(&minmax[2 * b],     __float_as_int(vmin));
        atomicMax(&minmax[2 * b + 1], __float_as_int(vmax));
    }
}

// ---------------------------------------------------------------------------
// Pass 2: per-image histogram, LDS-privatized. grid = (64, B) x 256 threads.
__global__ __launch_bounds__(256)
void hist_kernel(const float* __restrict__ blur, const int* __restrict__ minmax,
                 unsigned* __restrict__ hist) {
    __shared__ unsigned h[NBINS];
    const int b = blockIdx.y;
    h[threadIdx.x] = 0u;
    __syncthreads();
    float lo = __int_as_float(minmax[2 * b]);
    float hi = __int_as_float(minmax[2 * b + 1]);
    float width = (hi - lo) / (float)NBINS;
    float inv = 1.0f / fmaxf(width, 1e-12f);
    const float* img = blur + (size_t)b * NPIX;
    for (int i = blockIdx.x * blockDim.x + threadIdx.x; i < NPIX;
         i += gridDim.x * blockDim.x) {
        int idx = (int)floorf((img[i] - lo) * inv);
        idx = min(max(idx, 0), NBINS - 1);
        atomicAdd(&h[idx], 1u);
    }
    __syncthreads();
    atomicAdd(&hist[b * NBINS + threadIdx.x], h[threadIdx.x]);
}

// ---------------------------------------------------------------------------
// Pass 3: Otsu scan, one 256-thread block per image.
__global__ __launch_bounds__(256)
void otsu_kernel(const unsigned* __restrict__ hist,
                 const int* __restrict__ minmax, float* __restrict__ thresh) {
    const int b = blockIdx.x;
    const int t = threadIdx.x;
    __shared__ float cnt[NBINS], w1[NBINS], cs[NBINS];
    __shared__ float bv[NBINS];
    __shared__ int bi[NBINS];

    float lo = __int_as_float(minmax[2 * b]);
    float hi = __int_as_float(minmax[2 * b + 1]);
    float width = (hi - lo) / (float)NBINS;

    float c = (float)hist[b * NBINS + t];
    float center = lo + ((float)t + 0.5f) * width;
    cnt[t] = c;
    w1[t] = c;
    cs[t] = c * center;
    __syncthreads();

    // Hillis-Steele inclusive scans of counts and counts*centers.
    for (int off = 1; off < NBINS; off <<= 1) {
        float aw = (t >= off) ? w1[t - off] : 0.0f;
        float ac = (t >= off) ? cs[t - off] : 0.0f;
        __syncthreads();
        w1[t] += aw;
        cs[t] += ac;
        __syncthreads();
    }
    float Wt = w1[NBINS - 1], CSt = cs[NBINS - 1];

    // Suffix sums at t+1 follow from totals: w2[t+1]=Wt-w1[t], rcs=CSt-cs[t].
    if (t < NBINS - 1) {
        float w1i = w1[t], w2i = Wt - w1[t];
        float m1 = cs[t] / fmaxf(w1i, 1e-12f);
        float m2 = (CSt - cs[t]) / fmaxf(w2i, 1e-12f);
        float d = m1 - m2;
        bv[t] = w1i * w2i * d * d;
    } else {
        bv[t] = -3.4e38f;
    }
    bi[t] = t;
    __syncthreads();

    // argmax, first-occurrence on ties (matches jnp.argmax).
    for (int s = NBINS / 2; s > 0; s >>= 1) {
        if (t < s) {
            if (bv[t + s] > bv[t] || (bv[t + s] == bv[t] && bi[t + s] < bi[t])) {
                bv[t] = bv[t + s];
                bi[t] = bi[t + s];
            }
        }
        __syncthreads();
    }
    if (t == 0) thresh[b] = lo + ((float)bi[0] + 0.5f) * width;
}

// ---------------------------------------------------------------------------
// Pass 4: composite. grid = (1024, B) x 256 threads; one pixel, 3 channels.
__global__ __launch_bounds__(256)
void composite_kernel(const float* __restrict__ x,
                      const float* __restrict__ blur,
                      const float* __restrict__ thresh,
                      float* __restrict__ out) {
    const int b = blockIdx.y;
    const int p = blockIdx.x * blockDim.x + threadIdx.x;   // < NPIX exactly
    float t = thresh[b];
    float v = blur[(size_t)b * NPIX + p];
    float alpha = (v > t) ? 1.0f : v;
    float one_m = 1.0f - alpha;
    const float* xb = x + (size_t)b * 4 * NPIX;
    float* ob = out + (size_t)b * 3 * NPIX;
#pragma unroll
    for (int ch = 0; ch < 3; ++ch) {
        float r = xb[(size_t)(ch + 1) * NPIX + p];
        ob[(size_t)ch * NPIX + p] =
            fminf(fmaxf(alpha * r + one_m, 0.0f), 1.0f);
    }
}

// ---------------------------------------------------------------------------
extern "C" void kernel_launch(void* const* d_in, const int* in_sizes, int n_in,
                              void* d_out, int out_size, void* d_ws,
                              size_t ws_size, hipStream_t stream) {
    const float* x = (const float*)d_in[0];         // (32,4,512,512) f32
    float* out = (float*)d_out;                     // (32,3,512,512) f32

    // Workspace layout (needs 33.5 MB + 33 KB):
    char* ws = (char*)d_ws;
    float*    blur   = (float*)ws;                              // 32*512*512 f32
    size_t    off    = (size_t)B * NPIX * sizeof(float);
    int*      minmax = (int*)(ws + off);                        // 32*2
    unsigned* hist   = (unsigned*)(ws + off + 256);             // 32*256
    float*    thresh = (float*)(ws + off + 256 + B * NBINS * 4);// 32

    init_kernel<<<32, 256, 0, stream>>>(minmax, hist, thresh);
    blur_wmma_kernel<<<B * (H_IMG / 16) * (W_IMG / 16), 32, 0, stream>>>(
        x, blur, minmax);
    hist_kernel<<<dim3(64, B), 256, 0, stream>>>(blur, minmax, hist);
    otsu_kernel<<<B, 256, 0, stream>>>(hist, minmax, thresh);
    composite_kernel<<<dim3(NPIX / 256, B), 256, 0, stream>>>(
        x, blur, thresh, out);
}